// ResBlock_21921513079168
// MI455X (gfx1250) — compile-verified
//
#include <hip/hip_runtime.h>

// ---------------- problem constants ----------------
#define B_    2
#define NVV   163842
#define NFF   327680
#define INC   64
#define NECK  32
#define OUTC  128
#define EPSBN 1e-5f

typedef float v2f __attribute__((ext_vector_type(2)));
typedef float v8f __attribute__((ext_vector_type(8)));

// D = A(16x4 f32) * B(4x16 f32) + C(16x16 f32)
static __device__ inline v8f wmma4(v2f a, v2f b, v8f c) {
    return __builtin_amdgcn_wmma_f32_16x16x4_f32(
        /*neg_a=*/false, a, /*neg_b=*/false, b,
        /*c_mod=*/(short)0, c, /*reuse_a=*/false, /*reuse_b=*/false);
}

static __device__ inline v8f zero8() {
    v8f z; 
    #pragma unroll
    for (int e = 0; e < 8; ++e) z[e] = 0.0f;
    return z;
}

// ---------------------------------------------------------------
// K1: fold vals_G with EW/NS:  wEW[f][j] = sum_i vals_G[f,i,j]*EW[f,i]
// ---------------------------------------------------------------
__global__ void facew_kernel(const float* __restrict__ vals_G,
                             const float* __restrict__ EW,
                             const float* __restrict__ NS,
                             float* __restrict__ wEW, float* __restrict__ wNS) {
    long i = (long)blockIdx.x * blockDim.x + threadIdx.x;
    if (i >= (long)NFF * 3) return;
    int f = (int)(i / 3), j = (int)(i % 3);
    float se = 0.f, sn = 0.f;
    #pragma unroll
    for (int ii = 0; ii < 3; ++ii) {
        float v = vals_G[f * 9 + ii * 3 + j];
        se += v * EW[f * 3 + ii];
        sn += v * NS[f * 3 + ii];
    }
    wEW[i] = se; wNS[i] = sn;
}

// ---------------------------------------------------------------
// K2: fused conv1 (32x64) + shortcut conv (128x64) over x (B,64,NV).
// y1 -> (B,NV,32) channel-last; ysc -> (B,128,NV) channel-major.
// Accumulates BN1 / BNsc sums & sumsq.
// One wave = one 16-vertex tile; WMMA f32 16x16x4, K=64.
// ---------------------------------------------------------------
__global__ void conv1_sc_kernel(const float* __restrict__ x,
                                const float* __restrict__ w1, const float* __restrict__ b1,
                                const float* __restrict__ wsc, const float* __restrict__ bsc,
                                float* __restrict__ y1, float* __restrict__ ysc,
                                float* __restrict__ s1s, float* __restrict__ s1q,
                                float* __restrict__ scs, float* __restrict__ scq) {
    __shared__ float l1s[32], l1q[32], lcs[128], lcq[128];
    int tid = threadIdx.x;
    for (int i = tid; i < 32; i += blockDim.x) { l1s[i] = 0.f; l1q[i] = 0.f; }
    for (int i = tid; i < 128; i += blockDim.x) { lcs[i] = 0.f; lcq[i] = 0.f; }
    __syncthreads();

    const int lane  = tid & 31;
    const int wave  = tid >> 5;
    const int ncol  = lane & 15;   // N (B-frag) / M (A-frag) index within tile
    const int khalf = lane >> 4;   // K pair selector
    const int tilesPerB = (NVV + 15) / 16;
    long tile = (long)blockIdx.x * (blockDim.x >> 5) + wave;
    const long T = (long)B_ * tilesPerB;

    if (tile < T) {
        int b  = (int)(tile / tilesPerB);
        int n0 = (int)(tile % tilesPerB) * 16;
        int n  = n0 + ncol;
        int nc = n < NVV ? n : NVV - 1;
        const float* xb = x + (long)b * INC * NVV;

        v8f acc1[2], accs[8];
        #pragma unroll
        for (int i = 0; i < 2; ++i) acc1[i] = zero8();
        #pragma unroll
        for (int i = 0; i < 8; ++i) accs[i] = zero8();

        for (int k0 = 0; k0 < INC; k0 += 4) {
            int kk = k0 + 2 * khalf;
            v2f bf;
            bf.x = xb[(long)kk * NVV + nc];
            bf.y = xb[(long)(kk + 1) * NVV + nc];
            #pragma unroll
            for (int mt = 0; mt < 2; ++mt) {
                int m = mt * 16 + ncol;
                v2f af; af.x = w1[m * INC + kk]; af.y = w1[m * INC + kk + 1];
                acc1[mt] = wmma4(af, bf, acc1[mt]);
            }
            #pragma unroll
            for (int mt = 0; mt < 8; ++mt) {
                int m = mt * 16 + ncol;
                v2f af; af.x = wsc[m * INC + kk]; af.y = wsc[m * INC + kk + 1];
                accs[mt] = wmma4(af, bf, accs[mt]);
            }
        }

        bool valid = (n < NVV);
        float* y1b  = y1  + (long)b * NVV * NECK;
        float* yscb = ysc + (long)b * OUTC * NVV;
        #pragma unroll
        for (int mt = 0; mt < 2; ++mt)
            #pragma unroll
            for (int r = 0; r < 8; ++r) {
                int m = mt * 16 + khalf * 8 + r;
                float v = acc1[mt][r] + b1[m];
                if (valid) {
                    y1b[(long)n * NECK + m] = v;
                    atomicAdd(&l1s[m], v);
                    atomicAdd(&l1q[m], v * v);
                }
            }
        #pragma unroll
        for (int mt = 0; mt < 8; ++mt)
            #pragma unroll
            for (int r = 0; r < 8; ++r) {
                int m = mt * 16 + khalf * 8 + r;
                float v = accs[mt][r] + bsc[m];
                if (valid) {
                    yscb[(long)m * NVV + n] = v;
                    atomicAdd(&lcs[m], v);
                    atomicAdd(&lcq[m], v * v);
                }
            }
    }
    __syncthreads();
    for (int i = tid; i < 32; i += blockDim.x) { atomicAdd(&s1s[i], l1s[i]); atomicAdd(&s1q[i], l1q[i]); }
    for (int i = tid; i < 128; i += blockDim.x) { atomicAdd(&scs[i], lcs[i]); atomicAdd(&scq[i], lcq[i]); }
}

// ---------------------------------------------------------------
// BN finalize: (sum,sumsq) -> affine a,b   (one block, C threads)
// ---------------------------------------------------------------
__global__ void finalize_kernel(const float* __restrict__ sum, const float* __restrict__ sq,
                                const float* __restrict__ gamma, const float* __restrict__ beta,
                                float* __restrict__ a, float* __restrict__ b, float count) {
    int c = threadIdx.x;
    float mean = sum[c] / count;
    float var  = sq[c] / count - mean * mean;
    float inv  = rsqrtf(var + EPSBN);
    float av   = gamma[c] * inv;
    a[c] = av;
    b[c] = beta[c] - av * mean;
}

// ---------------------------------------------------------------
// In-place BN + ReLU on a channel-last (.., 32) buffer
// ---------------------------------------------------------------
__global__ void bnrelu32_kernel(float* __restrict__ buf, const float* __restrict__ a,
                                const float* __restrict__ b, long n) {
    long i = (long)blockIdx.x * blockDim.x + threadIdx.x;
    if (i >= n) return;
    int c = (int)(i & 31);
    float v = a[c] * buf[i] + b[c];
    buf[i] = v > 0.f ? v : 0.f;
}

// ---------------------------------------------------------------
// K3: per-face tangent gradients  g_ew/g_ns (B,NF,32) from h (B,NV,32)
// lane = channel -> every gather is one contiguous 128B line.
// ---------------------------------------------------------------
__global__ void facegrad_kernel(const float* __restrict__ h, const int* __restrict__ idx_G,
                                const float* __restrict__ wEW, const float* __restrict__ wNS,
                                float* __restrict__ gew, float* __restrict__ gns) {
    long t = (long)blockIdx.x * blockDim.x + threadIdx.x;
    long total = (long)B_ * NFF * NECK;
    if (t >= total) return;
    int c   = (int)(t & 31);
    long fb = t >> 5;
    int b = (int)(fb / NFF), f = (int)(fb % NFF);
    const float* hb = h + (long)b * NVV * NECK;
    float ae = 0.f, an = 0.f;
    #pragma unroll
    for (int j = 0; j < 3; ++j) {
        int ix = idx_G[f * 3 + j];
        float hv = hb[(long)ix * NECK + c];
        ae += wEW[f * 3 + j] * hv;
        an += wNS[f * 3 + j] * hv;
    }
    gew[t] = ae;
    gns[t] = an;
}

// ---------------------------------------------------------------
// K4: mesh-conv. Phase 1 (lane=channel): build 128-wide feature
// vector [x, lap, gv_ew, gv_ns] per vertex into LDS. Phase 2: WMMA
// combine with coeffs (32 x 128) -> mcout (B,NV,32).
// 128 threads/block = 4 waves; each wave owns a 16-vertex tile.
// ---------------------------------------------------------------
__global__ void meshconv_kernel(const float* __restrict__ h,
                                const float* __restrict__ gew, const float* __restrict__ gns,
                                const int* __restrict__ idx_L, const float* __restrict__ vals_L,
                                const int* __restrict__ idx_F2V, const float* __restrict__ vals_F2V,
                                const float* __restrict__ coeffs, const float* __restrict__ bias,
                                float* __restrict__ mcout) {
    __shared__ float feat[4][16][128];   // 32 KB
    int tid = threadIdx.x, lane = tid & 31, wave = tid >> 5;
    const int tilesPerB = (NVV + 15) / 16;
    long tile = (long)blockIdx.x * 4 + wave;
    const long T = (long)B_ * tilesPerB;
    bool active = tile < T;
    int b = 0, n0 = 0;
    if (active) { b = (int)(tile / tilesPerB); n0 = (int)(tile % tilesPerB) * 16; }

    if (active) {
        const float* hb = h   + (long)b * NVV * NECK;
        const float* ge = gew + (long)b * NFF * NECK;
        const float* gn = gns + (long)b * NFF * NECK;
        for (int v = 0; v < 16; ++v) {
            int n = n0 + v;
            float f0 = 0.f, f1 = 0.f, f2 = 0.f, f3 = 0.f;
            if (n < NVV) {
                f0 = hb[(long)n * NECK + lane];
                #pragma unroll
                for (int k = 0; k < 7; ++k) {
                    int ix = idx_L[n * 7 + k];
                    f1 += vals_L[n * 7 + k] * hb[(long)ix * NECK + lane];
                }
                #pragma unroll
                for (int k = 0; k < 6; ++k) {
                    int ix = idx_F2V[n * 6 + k];
                    float w = vals_F2V[n * 6 + k];
                    f2 += w * ge[(long)ix * NECK + lane];
                    f3 += w * gn[(long)ix * NECK + lane];
                }
            }
            // k-index = c*4 + tap  (matches coeffs[o][c][t] flattening)
            float4* row = (float4*)&feat[wave][v][0];
            row[lane] = make_float4(f0, f1, f2, f3);
        }
    }
    __syncthreads();
    if (active) {
        int ncol = lane & 15, khalf = lane >> 4;
        v8f acc[2];
        acc[0] = zero8(); acc[1] = zero8();
        for (int k0 = 0; k0 < 128; k0 += 4) {
            int kk = k0 + 2 * khalf;
            v2f bf;
            bf.x = feat[wave][ncol][kk];
            bf.y = feat[wave][ncol][kk + 1];
            #pragma unroll
            for (int mt = 0; mt < 2; ++mt) {
                int m = mt * 16 + ncol;
                v2f af; af.x = coeffs[m * 128 + kk]; af.y = coeffs[m * 128 + kk + 1];
                acc[mt] = wmma4(af, bf, acc[mt]);
            }
        }
        int n = n0 + ncol;
        if (n < NVV) {
            float* ob = mcout + (long)b * NVV * NECK;
            #pragma unroll
            for (int mt = 0; mt < 2; ++mt)
                #pragma unroll
                for (int r = 0; r < 8; ++r) {
                    int m = mt * 16 + khalf * 8 + r;
                    ob[(long)n * NECK + m] = acc[mt][r] + bias[m];
                }
        }
    }
}

// ---------------------------------------------------------------
// per-channel stats over a channel-last (rows, 32) buffer
// ---------------------------------------------------------------
__global__ void stats32_kernel(const float* __restrict__ buf, long rows,
                               float* __restrict__ ssum, float* __restrict__ ssq) {
    __shared__ float s[32], q[32];
    int tid = threadIdx.x;
    if (tid < 32) { s[tid] = 0.f; q[tid] = 0.f; }
    __syncthreads();
    int lane = tid & 31;
    long r0 = (long)blockIdx.x * (blockDim.x >> 5) + (tid >> 5);
    long rs = (long)gridDim.x * (blockDim.x >> 5);
    float ls = 0.f, lq = 0.f;
    for (long r = r0; r < rows; r += rs) {
        float v = buf[r * 32 + lane];
        ls += v; lq += v * v;
    }
    atomicAdd(&s[lane], ls);
    atomicAdd(&q[lane], lq);
    __syncthreads();
    if (tid < 32) { atomicAdd(&ssum[tid], s[tid]); atomicAdd(&ssq[tid], q[tid]); }
}

// ---------------------------------------------------------------
// K5: conv3 (128x32) over h2 (B,NV,32) -> y3 (B,128,NV) + BN3 stats
// ---------------------------------------------------------------
__global__ void conv3_kernel(const float* __restrict__ h2,
                             const float* __restrict__ w3, const float* __restrict__ b3,
                             float* __restrict__ y3,
                             float* __restrict__ s3s, float* __restrict__ s3q) {
    __shared__ float ls[128], lq[128];
    int tid = threadIdx.x;
    for (int i = tid; i < 128; i += blockDim.x) { ls[i] = 0.f; lq[i] = 0.f; }
    __syncthreads();

    const int lane = tid & 31, wave = tid >> 5;
    const int ncol = lane & 15, khalf = lane >> 4;
    const int tilesPerB = (NVV + 15) / 16;
    long tile = (long)blockIdx.x * (blockDim.x >> 5) + wave;
    const long T = (long)B_ * tilesPerB;

    if (tile < T) {
        int b  = (int)(tile / tilesPerB);
        int n0 = (int)(tile % tilesPerB) * 16;
        int n  = n0 + ncol;
        int nc = n < NVV ? n : NVV - 1;
        const float* hb = h2 + (long)b * NVV * NECK;

        v8f acc[8];
        #pragma unroll
        for (int i = 0; i < 8; ++i) acc[i] = zero8();

        for (int k0 = 0; k0 < NECK; k0 += 4) {
            int kk = k0 + 2 * khalf;
            v2f bf;
            bf.x = hb[(long)nc * NECK + kk];
            bf.y = hb[(long)nc * NECK + kk + 1];
            #pragma unroll
            for (int mt = 0; mt < 8; ++mt) {
                int m = mt * 16 + ncol;
                v2f af; af.x = w3[m * NECK + kk]; af.y = w3[m * NECK + kk + 1];
                acc[mt] = wmma4(af, bf, acc[mt]);
            }
        }

        bool valid = (n < NVV);
        float* yb = y3 + (long)b * OUTC * NVV;
        #pragma unroll
        for (int mt = 0; mt < 8; ++mt)
            #pragma unroll
            for (int r = 0; r < 8; ++r) {
                int m = mt * 16 + khalf * 8 + r;
                float v = acc[mt][r] + b3[m];
                if (valid) {
                    yb[(long)m * NVV + n] = v;
                    atomicAdd(&ls[m], v);
                    atomicAdd(&lq[m], v * v);
                }
            }
    }
    __syncthreads();
    for (int i = tid; i < 128; i += blockDim.x) { atomicAdd(&s3s[i], ls[i]); atomicAdd(&s3q[i], lq[i]); }
}

// ---------------------------------------------------------------
// K6: final residual:  out = relu(bn3(y3) + bnsc(ysc)),  (B,128,NV)
// grid.y encodes (b*128 + o) -> no per-thread division.
// ---------------------------------------------------------------
__global__ void final_kernel(const float* __restrict__ y3, const float* __restrict__ ysc,
                             const float* __restrict__ a3, const float* __restrict__ b3,
                             const float* __restrict__ asc, const float* __restrict__ bsc,
                             float* __restrict__ out) {
    int n = blockIdx.x * blockDim.x + threadIdx.x;
    if (n >= NVV) return;
    int o = blockIdx.y & (OUTC - 1);
    long i = (long)blockIdx.y * NVV + n;
    float v = a3[o] * y3[i] + b3[o] + asc[o] * ysc[i] + bsc[o];
    out[i] = v > 0.f ? v : 0.f;
}

// ---------------------------------------------------------------
extern "C" void kernel_launch(void* const* d_in, const int* in_sizes, int n_in,
                              void* d_out, int out_size, void* d_ws, size_t ws_size,
                              hipStream_t stream) {
    (void)in_sizes; (void)n_in; (void)out_size; (void)ws_size;
    const float* x       = (const float*)d_in[0];
    const float* w1      = (const float*)d_in[1];
    const float* b1      = (const float*)d_in[2];
    const float* gamma1  = (const float*)d_in[3];
    const float* beta1   = (const float*)d_in[4];
    const float* coeffs  = (const float*)d_in[5];
    const float* mc_bias = (const float*)d_in[6];
    const float* gamma2  = (const float*)d_in[7];
    const float* beta2   = (const float*)d_in[8];
    const float* w3      = (const float*)d_in[9];
    const float* b3      = (const float*)d_in[10];
    const float* gamma3  = (const float*)d_in[11];
    const float* beta3   = (const float*)d_in[12];
    const float* wsc     = (const float*)d_in[13];
    const float* bsc     = (const float*)d_in[14];
    const float* gammasc = (const float*)d_in[15];
    const float* betasc  = (const float*)d_in[16];
    const int*   idx_G   = (const int*)  d_in[17];
    const float* vals_G  = (const float*)d_in[18];
    const float* EW      = (const float*)d_in[19];
    const float* NS      = (const float*)d_in[20];
    const int*   idx_L   = (const int*)  d_in[21];
    const float* vals_L  = (const float*)d_in[22];
    const int*   idx_F2V = (const int*)  d_in[23];
    const float* vals_F2V= (const float*)d_in[24];
    float* out = (float*)d_out;

    // workspace layout (floats)
    const long SZ_H   = (long)B_ * NVV * NECK;    // 10,485,888
    const long SZ_YSC = (long)B_ * OUTC * NVV;    // 41,943,552
    const long SZ_G1  = (long)B_ * NFF * NECK;    // 20,971,520
    float* ws   = (float*)d_ws;
    float* h    = ws;                 // conv1 out -> bn1+relu in place
    float* mc   = h + SZ_H;           // meshconv out -> bn2+relu in place
    float* ysc  = mc + SZ_H;          // raw shortcut conv
    float* greg = ysc + SZ_YSC;       // gew|gns, later reused for y3
    float* gew  = greg;
    float* gns  = greg + SZ_G1;
    float* y3   = greg;               // reuse after meshconv consumed gew/gns
    float* wEWp = greg + SZ_YSC;
    float* wNSp = wEWp + (long)NFF * 3;
    float* st   = wNSp + (long)NFF * 3;
    // stats sums
    float *s1s = st,       *s1q = st + 32;
    float *scs = st + 64,  *scq = st + 192;
    float *s2s = st + 320, *s2q = st + 352;
    float *s3s = st + 384, *s3q = st + 512;
    // affine params
    float* AB = st + 640;
    float *a1 = AB,        *bb1 = AB + 32;
    float *asc= AB + 64,   *bbsc= AB + 192;
    float *a2 = AB + 320,  *bb2 = AB + 352;
    float *a3 = AB + 384,  *bb3 = AB + 512;

    const float count = (float)((long)B_ * NVV);
    const int tilesPerB = (NVV + 15) / 16;
    const long Tt = (long)B_ * tilesPerB;                  // 20482 tiles
    const int convBlocks = (int)((Tt + 7) / 8);            // 8 waves/block
    const int mcBlocks   = (int)((Tt + 3) / 4);            // 4 waves/block
    const long hElems = SZ_H;

    hipMemsetAsync(st, 0, 640 * sizeof(float), stream);

    facew_kernel<<<(NFF * 3 + 255) / 256, 256, 0, stream>>>(vals_G, EW, NS, wEWp, wNSp);

    conv1_sc_kernel<<<convBlocks, 256, 0, stream>>>(x, w1, b1, wsc, bsc, h, ysc,
                                                    s1s, s1q, scs, scq);

    finalize_kernel<<<1, 32, 0, stream>>>(s1s, s1q, gamma1, beta1, a1, bb1, count);
    finalize_kernel<<<1, 128, 0, stream>>>(scs, scq, gammasc, betasc, asc, bbsc, count);

    bnrelu32_kernel<<<(int)((hElems + 255) / 256), 256, 0, stream>>>(h, a1, bb1, hElems);

    facegrad_kernel<<<(int)(((long)B_ * NFF * NECK + 255) / 256), 256, 0, stream>>>(
        h, idx_G, wEWp, wNSp, gew, gns);

    meshconv_kernel<<<mcBlocks, 128, 0, stream>>>(h, gew, gns, idx_L, vals_L,
                                                  idx_F2V, vals_F2V, coeffs, mc_bias, mc);

    stats32_kernel<<<512, 256, 0, stream>>>(mc, (long)B_ * NVV, s2s, s2q);
    finalize_kernel<<<1, 32, 0, stream>>>(s2s, s2q, gamma2, beta2, a2, bb2, count);

    bnrelu32_kernel<<<(int)((hElems + 255) / 256), 256, 0, stream>>>(mc, a2, bb2, hElems);

    conv3_kernel<<<convBlocks, 256, 0, stream>>>(mc, w3, b3, y3, s3s, s3q);
    finalize_kernel<<<1, 128, 0, stream>>>(s3s, s3q, gamma3, beta3, a3, bb3, count);

    dim3 fgrid((NVV + 255) / 256, B_ * OUTC);
    final_kernel<<<fgrid, 256, 0, stream>>>(y3, ysc, a3, bb3, asc, bbsc, out);
}